// DSI_82059645157485
// MI455X (gfx1250) — compile-verified
//
#include <hip/hip_runtime.h>
#include <math.h>

// Problem constants (reference: B=64, S=2048, D=256, N_HOP=3)
#define B_DIM 64
#define S_DIM 2048
#define D_DIM 256
#define N_HOP 3
#define EPS 1e-8f

typedef __attribute__((ext_vector_type(2))) float v2f;
typedef __attribute__((ext_vector_type(8))) float v8f;

// ---------------------------------------------------------------------------
// K1: scores[b,s] = (q . s) / (max(|q|,eps) * max(|s|,eps))
// grid = (B, S/8), block = 256 (8 waves, one state row per wave)
// ---------------------------------------------------------------------------
__global__ __launch_bounds__(256) void k_scores(const float* __restrict__ state,
                                                const float* __restrict__ q,
                                                float* __restrict__ scores) {
    __shared__ float qs[D_DIM];
    const int b = blockIdx.x;
    const int tid = threadIdx.x;
    qs[tid] = q[b * D_DIM + tid];
    __syncthreads();

    const int wave = tid >> 5;
    const int lane = tid & 31;

    // |q| (each wave computes it redundantly; D=256 -> 8 elems/lane)
    float qn2 = 0.f;
#pragma unroll
    for (int k = 0; k < 8; ++k) {
        float v = qs[lane + 32 * k];
        qn2 += v * v;
    }
#pragma unroll
    for (int off = 16; off > 0; off >>= 1) qn2 += __shfl_xor(qn2, off, 32);
    const float qn = fmaxf(sqrtf(qn2), EPS);

    const int s = blockIdx.y * 8 + wave;
    const float* row = state + ((size_t)b * S_DIM + s) * D_DIM;
    float dot = 0.f, sn2 = 0.f;
#pragma unroll
    for (int k = 0; k < 8; ++k) {
        float v = row[lane + 32 * k];
        dot += v * qs[lane + 32 * k];
        sn2 += v * v;
    }
#pragma unroll
    for (int off = 16; off > 0; off >>= 1) {
        dot += __shfl_xor(dot, off, 32);
        sn2 += __shfl_xor(sn2, off, 32);
    }
    if (lane == 0) {
        const float sn = fmaxf(sqrtf(sn2), EPS);
        scores[b * S_DIM + s] = dot / (qn * sn);
    }
}

// ---------------------------------------------------------------------------
// K2: dist[b,:] = softmax(scores[b,:]) over S=2048. grid = B, block = 256.
// ---------------------------------------------------------------------------
__global__ __launch_bounds__(256) void k_softmax(const float* __restrict__ scores,
                                                 float* __restrict__ dist) {
    __shared__ float red[256];
    const int b = blockIdx.x;
    const int tid = threadIdx.x;
    const float* in = scores + (size_t)b * S_DIM;

    float v[8];
    float m = -INFINITY;
#pragma unroll
    for (int k = 0; k < 8; ++k) {
        v[k] = in[tid + 256 * k];
        m = fmaxf(m, v[k]);
    }
    red[tid] = m;
    __syncthreads();
    for (int off = 128; off > 0; off >>= 1) {
        if (tid < off) red[tid] = fmaxf(red[tid], red[tid + off]);
        __syncthreads();
    }
    m = red[0];
    __syncthreads();

    float sum = 0.f;
#pragma unroll
    for (int k = 0; k < 8; ++k) {
        v[k] = __expf(v[k] - m);
        sum += v[k];
    }
    red[tid] = sum;
    __syncthreads();
    for (int off = 128; off > 0; off >>= 1) {
        if (tid < off) red[tid] += red[tid + off];
        __syncthreads();
    }
    const float inv = 1.f / red[0];
    float* out = dist + (size_t)b * S_DIM;
#pragma unroll
    for (int k = 0; k < 8; ++k) out[tid + 256 * k] = v[k] * inv;
}

// ---------------------------------------------------------------------------
// K3: in-place  state[g0+0..31, :] = rows @ Wh^T + bh     (fp32 WMMA)
//     fused     qacc[b, e] += sum_m dist[b, s0+m] * new_state[m, e]
// grid = B*S/32, block = 256 (8 waves).
// Wave register tile: 32 rows x 32 cols = four 16x16 accumulators.
// Per K-step (K=4): 2 A-frags (LDS) + 2 B-frags (global) -> 4 WMMAs.
// ---------------------------------------------------------------------------
#define ROWS_PER_BLOCK 32
#define LDS_STRIDE 260  // 256 + 4 pad: breaks 16-way bank conflict on K-column reads

__global__ __launch_bounds__(256) void k_gemm_qacc(float* __restrict__ state,
                                                   const float* __restrict__ Wh,
                                                   const float* __restrict__ bh,
                                                   const float* __restrict__ dist,
                                                   float* __restrict__ qacc) {
    __shared__ float rows[ROWS_PER_BLOCK * LDS_STRIDE];
    __shared__ float dloc[ROWS_PER_BLOCK];

    const int tid = threadIdx.x;
    const size_t g0 = (size_t)blockIdx.x * ROWS_PER_BLOCK;  // rows never span b
    const int b = (int)(g0 / S_DIM);
    const int s0 = (int)(g0 % S_DIM);

    const int wave = tid >> 5;
    const int lane = tid & 31;
    const int half = lane >> 4;  // lanes 16-31: K=k+2,k+3 (A/B frags), rows M+8 (C)
    const int l15 = lane & 15;
    const int e0 = wave * 32;    // this wave's 32-column stripe

    // Warm this wave's W stripe into the cache hierarchy (global_prefetch_b8).
    {
        const float* wrow = Wh + (size_t)(e0 + (lane & 31)) * D_DIM;
        __builtin_prefetch(wrow, 0, 3);
        __builtin_prefetch(wrow + 128, 0, 3);
    }

    // Stage the 32x256 input rows into LDS (float4, fully coalesced).
#pragma unroll
    for (int i = 0; i < 8; ++i) {
        const int f = i * 256 + tid;   // float4 index; 64 float4 per row
        const int r = f >> 6;
        const int c4 = f & 63;
        float4 t = *(const float4*)(state + (g0 + r) * D_DIM + c4 * 4);
        float* dst = rows + r * LDS_STRIDE + c4 * 4;
        dst[0] = t.x; dst[1] = t.y; dst[2] = t.z; dst[3] = t.w;
    }
    if (tid < ROWS_PER_BLOCK) dloc[tid] = dist[(size_t)b * S_DIM + s0 + tid];
    __syncthreads();

    // A-frag sources: row-tile 0 = rows 0..15, row-tile 1 = rows 16..31.
    const float* abase0 = rows + l15 * LDS_STRIDE + 2 * half;
    const float* abase1 = rows + (16 + l15) * LDS_STRIDE + 2 * half;
    // B-frag sources: B[k][n] = W[e+n][k]  (y = x @ W^T), col-tiles 0 and 1.
    const float* wbase0 = Wh + (size_t)(e0 + l15) * D_DIM + 2 * half;
    const float* wbase1 = Wh + (size_t)(e0 + 16 + l15) * D_DIM + 2 * half;

    v8f c00 = {}, c01 = {}, c10 = {}, c11 = {};
    for (int kk = 0; kk < D_DIM; kk += 4) {
        v2f a0, a1, b0, b1;
        a0.x = abase0[kk]; a0.y = abase0[kk + 1];
        a1.x = abase1[kk]; a1.y = abase1[kk + 1];
        b0.x = wbase0[kk]; b0.y = wbase0[kk + 1];
        b1.x = wbase1[kk]; b1.y = wbase1[kk + 1];
        c00 = __builtin_amdgcn_wmma_f32_16x16x4_f32(false, a0, false, b0, (short)0, c00, false, false);
        c01 = __builtin_amdgcn_wmma_f32_16x16x4_f32(false, a0, false, b1, (short)0, c01, false, false);
        c10 = __builtin_amdgcn_wmma_f32_16x16x4_f32(false, a1, false, b0, (short)0, c10, false, false);
        c11 = __builtin_amdgcn_wmma_f32_16x16x4_f32(false, a1, false, b1, (short)0, c11, false, false);
    }

    // Epilogue: bias add, in-place writeback, fused dist-weighted q accumulation.
#pragma unroll
    for (int ct = 0; ct < 2; ++ct) {
        const int e = e0 + ct * 16 + l15;
        const float bias = bh[e];
        float qpart = 0.f;
#pragma unroll
        for (int rt = 0; rt < 2; ++rt) {
            const v8f& c = (ct == 0) ? (rt == 0 ? c00 : c10) : (rt == 0 ? c01 : c11);
#pragma unroll
            for (int r = 0; r < 8; ++r) {
                const int M = rt * 16 + r + half * 8;  // C layout: vgpr r -> row M / M+8
                const float val = c[r] + bias;
                state[(g0 + M) * D_DIM + e] = val;
                qpart += dloc[M] * val;
            }
        }
        qpart += __shfl_xor(qpart, 16, 32);  // lanes l and l+16 share column e
        if (half == 0) atomicAdd(&qacc[(size_t)b * D_DIM + e], qpart);
    }
}

// ---------------------------------------------------------------------------
// K4: q += qacc; qacc = 0.  grid = B, block = 256 (= D).
// ---------------------------------------------------------------------------
__global__ __launch_bounds__(256) void k_qupdate(float* __restrict__ q,
                                                 float* __restrict__ qacc) {
    const int i = blockIdx.x * D_DIM + threadIdx.x;
    q[i] += qacc[i];
    qacc[i] = 0.f;
}

// ---------------------------------------------------------------------------
extern "C" void kernel_launch(void* const* d_in, const int* in_sizes, int n_in,
                              void* d_out, int out_size, void* d_ws, size_t ws_size,
                              hipStream_t stream) {
    (void)in_sizes; (void)n_in; (void)out_size; (void)ws_size;

    const float* in_q     = (const float*)d_in[0];  // [B, D]
    const float* in_state = (const float*)d_in[1];  // [B, S, D]
    const float* Wall     = (const float*)d_in[2];  // [3, D, D]
    const float* ball     = (const float*)d_in[3];  // [3, D]

    const size_t stateBytes = (size_t)B_DIM * S_DIM * D_DIM * sizeof(float); // 134217728
    const size_t bsBytes    = (size_t)B_DIM * S_DIM * sizeof(float);         // 524288
    const size_t bdBytes    = (size_t)B_DIM * D_DIM * sizeof(float);         // 65536

    char* ws = (char*)d_ws;
    float* state  = (float*)(ws);
    float* scores = (float*)(ws + stateBytes);
    float* dist   = (float*)(ws + stateBytes + bsBytes);
    float* q      = (float*)(ws + stateBytes + 2 * bsBytes);
    float* qacc   = (float*)(ws + stateBytes + 2 * bsBytes + bdBytes);

    // Inputs must stay untouched: work on workspace copies.
    hipMemcpyAsync(state, in_state, stateBytes, hipMemcpyDeviceToDevice, stream);
    hipMemcpyAsync(q, in_q, bdBytes, hipMemcpyDeviceToDevice, stream);
    hipMemsetAsync(qacc, 0, bdBytes, stream);

    const dim3 gScores(B_DIM, S_DIM / 8);
    for (int h = 0; h < N_HOP; ++h) {
        k_scores<<<gScores, 256, 0, stream>>>(state, q, scores);
        k_softmax<<<B_DIM, 256, 0, stream>>>(scores, dist);
        k_gemm_qacc<<<(B_DIM * S_DIM) / ROWS_PER_BLOCK, 256, 0, stream>>>(
            state, Wall + (size_t)h * D_DIM * D_DIM, ball + (size_t)h * D_DIM,
            dist, qacc);
        k_qupdate<<<B_DIM, 256, 0, stream>>>(q, qacc);
    }
    // Final cosine softmax with updated q & state.
    k_scores<<<gScores, 256, 0, stream>>>(state, q, scores);
    k_softmax<<<B_DIM, 256, 0, stream>>>(scores, dist);

    // d_out = [query_emb (B*D) | final_dist (B*S)]
    float* outp = (float*)d_out;
    hipMemcpyAsync(outp, q, bdBytes, hipMemcpyDeviceToDevice, stream);
    hipMemcpyAsync(outp + B_DIM * D_DIM, dist, bsBytes, hipMemcpyDeviceToDevice, stream);
}